// DiscriminativeLoss_6614249636120
// MI455X (gfx1250) — compile-verified
//
#include <hip/hip_runtime.h>
#include <hip/hip_bf16.h>
#include <math.h>

// Problem constants (match reference setup_inputs / module constants)
#define BB 16
#define NN 65536
#define DD 32
#define KK 64
#define CHUNKS 32
#define PTS (NN / CHUNKS)   // 2048 points per block

#define DELTA_V 0.5f
#define TWO_DELTA_D 3.0f    // 2 * DELTA_D
#define GAMMA_W 0.001f

typedef __attribute__((ext_vector_type(2))) float v2f;
typedef __attribute__((ext_vector_type(8))) float v8f;

// Workspace layout (floats)
//   [0, 32768)          sums   -> overwritten in place with means
//   [32768, 33792)      counts (B*K)
//   [33792, 34816)      inst_pen (B*K)
//   [34816, 34820)      accum: varAcc, distAcc, regAcc, validCnt
#define WS_SUMS    0
#define WS_COUNTS  32768
#define WS_PEN     33792
#define WS_ACCUM   34816
#define WS_FLOATS  34820

__device__ __forceinline__ float waveSum32(float v) {
    #pragma unroll
    for (int o = 16; o > 0; o >>= 1) v += __shfl_xor(v, o, 32);
    return v;
}

// ---------------- kernel 0: zero workspace ----------------
__global__ void k_zero(float* ws) {
    int i = blockIdx.x * blockDim.x + threadIdx.x;
    if (i < WS_FLOATS) ws[i] = 0.0f;
}

// ---------------- kernel 1: segment sums + counts ----------------
// grid (BB, CHUNKS), block 256. Each 32-thread wave handles one point at a
// time: lane d reads emb[b][n][d] (fully coalesced 128B), LDS-atomic into
// sums[id*32+d] -> within a wave the 32 addresses hit 32 distinct banks.
__global__ void k_segsum(const float* __restrict__ emb,
                         const int* __restrict__ ids,
                         const unsigned char* __restrict__ mask,
                         float* __restrict__ sums,
                         float* __restrict__ counts) {
    __shared__ float sS[KK * DD];
    __shared__ float cS[KK];
    const int b = blockIdx.x;
    const int chunk = blockIdx.y;
    const int tid = threadIdx.x;

    for (int i = tid; i < KK * DD; i += 256) sS[i] = 0.0f;
    for (int i = tid; i < KK; i += 256) cS[i] = 0.0f;
    __syncthreads();

    const int group = tid >> 5;   // wave id within block (wave32)
    const int d = tid & 31;
    const int base = chunk * PTS;
    const size_t ebase = (size_t)b * NN * DD;

    for (int n = base + group; n < base + PTS; n += 8) {
        int id = ids[b * NN + n];
        bool valid = (mask[b * NN + n] != 0) && (id >= 0) && (id < KK);
        float v = emb[ebase + (size_t)n * DD + d];
        if (valid) {
            atomicAdd(&sS[id * DD + d], v);
            if (d == 0) atomicAdd(&cS[id], 1.0f);
        }
    }
    __syncthreads();

    for (int i = tid; i < KK * DD; i += 256)
        if (sS[i] != 0.0f) atomicAdd(&sums[b * KK * DD + i], sS[i]);
    for (int i = tid; i < KK; i += 256)
        if (cS[i] != 0.0f) atomicAdd(&counts[b * KK + i], cS[i]);
}

// ---------------- kernel 2: means in place ----------------
__global__ void k_means(float* __restrict__ sums, const float* __restrict__ counts) {
    int i = blockIdx.x * blockDim.x + threadIdx.x;
    if (i < BB * KK * DD) {
        float c = counts[i / DD];
        sums[i] = sums[i] / fmaxf(c, 1.0f);
    }
}

// ---------------- kernel 3: variance (pull) penalty ----------------
__global__ void k_var(const float* __restrict__ emb,
                      const int* __restrict__ ids,
                      const unsigned char* __restrict__ mask,
                      const float* __restrict__ means,
                      float* __restrict__ inst_pen) {
    __shared__ float pS[KK];
    const int b = blockIdx.x;
    const int chunk = blockIdx.y;
    const int tid = threadIdx.x;

    for (int i = tid; i < KK; i += 256) pS[i] = 0.0f;
    __syncthreads();

    const int group = tid >> 5;
    const int d = tid & 31;
    const int base = chunk * PTS;
    const size_t ebase = (size_t)b * NN * DD;

    for (int n = base + group; n < base + PTS; n += 8) {
        int id = ids[b * NN + n];
        bool valid = (mask[b * NN + n] != 0) && (id >= 0) && (id < KK);
        int idc = id < 0 ? 0 : (id >= KK ? KK - 1 : id);
        float v = emb[ebase + (size_t)n * DD + d];
        float m = means[(b * KK + idc) * DD + d];
        float diff = valid ? (v - m) : 0.0f;
        float sq = diff * diff;
        sq = waveSum32(sq);                 // per-point squared distance
        if (d == 0 && valid) {
            float dist = sqrtf(sq);
            float t = fmaxf(dist - DELTA_V, 0.0f);
            atomicAdd(&pS[idc], t * t);
        }
    }
    __syncthreads();
    for (int i = tid; i < KK; i += 256)
        if (pS[i] != 0.0f) atomicAdd(&inst_pen[b * KK + i], pS[i]);
}

// ---------------- kernel 4: per-batch losses, Gram via WMMA ----------------
// One wave (32 threads) per batch. G = M * M^T, M = (64 x 32) means.
// V_WMMA_F32_16X16X4_F32: A = 16x4 f32 (v2f), B = 4x16 f32 (v2f), C/D v8f.
__global__ void k_batch(const float* __restrict__ means,
                        const float* __restrict__ counts,
                        const float* __restrict__ inst_pen,
                        float* __restrict__ accum) {
    __shared__ float M[KK * DD];     // 8 KB
    __shared__ float G[KK * KK];     // 16 KB
    __shared__ float pres[KK];
    __shared__ float cnts[KK];

    const int b = blockIdx.x;
    const int lane = threadIdx.x;    // 0..31, single wave

    for (int i = lane; i < KK * DD; i += 32) M[i] = means[b * KK * DD + i];
    {
        float c0 = counts[b * KK + lane];
        float c1 = counts[b * KK + lane + 32];
        cnts[lane] = c0;       cnts[lane + 32] = c1;
        pres[lane] = c0 > 0.0f ? 1.0f : 0.0f;
        pres[lane + 32] = c1 > 0.0f ? 1.0f : 0.0f;
    }
    __syncthreads();

    const int r = lane & 15;          // row/col within a 16-tile
    const int khalf = (lane >> 4) * 2; // K sub-offset held by this half-wave

    #pragma unroll
    for (int ti = 0; ti < 4; ++ti) {
        // A fragments for row-tile ti, all 8 K-chunks (K=4 each)
        v2f aF[8];
        #pragma unroll
        for (int kc = 0; kc < 8; ++kc) {
            aF[kc].x = M[(ti * 16 + r) * DD + kc * 4 + khalf];
            aF[kc].y = M[(ti * 16 + r) * DD + kc * 4 + khalf + 1];
        }
        #pragma unroll
        for (int tj = 0; tj < 4; ++tj) {
            v8f c = {};
            #pragma unroll
            for (int kc = 0; kc < 8; ++kc) {
                v2f bF;   // B = M^T tile: B[k][n] = M[tj*16+n][k]
                bF.x = M[(tj * 16 + r) * DD + kc * 4 + khalf];
                bF.y = M[(tj * 16 + r) * DD + kc * 4 + khalf + 1];
                c = __builtin_amdgcn_wmma_f32_16x16x4_f32(
                        false, aF[kc], false, bF, (short)0, c, false, false);
            }
            // C/D layout: VGPR j, lanes 0-15 -> M=j, lanes 16-31 -> M=j+8
            const int half = lane >> 4;
            #pragma unroll
            for (int j = 0; j < 8; ++j)
                G[(ti * 16 + j + half * 8) * KK + tj * 16 + r] = c[j];
        }
    }
    __syncthreads();

    // instance count
    float p0 = pres[lane], p1 = pres[lane + 32];
    float ninst = waveSum32(p0 + p1);

    // variance (pull) loss
    float pen0 = inst_pen[b * KK + lane];
    float pen1 = inst_pen[b * KK + lane + 32];
    float var_num = waveSum32(pen0 / fmaxf(cnts[lane], 1.0f) * p0 +
                              pen1 / fmaxf(cnts[lane + 32], 1.0f) * p1);
    float var_b = var_num / fmaxf(ninst, 1.0f);

    // regularization from Gram diagonal
    float d0 = G[lane * KK + lane];
    float d1 = G[(lane + 32) * KK + lane + 32];
    float reg_num = waveSum32(sqrtf(fmaxf(d0, 0.0f)) * p0 +
                              sqrtf(fmaxf(d1, 0.0f)) * p1);
    float reg_b = reg_num / fmaxf(ninst, 1.0f);

    // push loss over upper-triangular pairs
    float psum = 0.0f;
    for (int i = 0; i < KK; ++i) {
        float gii = G[i * KK + i];
        float pi = pres[i];
        for (int j = i + 1 + lane; j < KK; j += 32) {
            float dsq = gii + G[j * KK + j] - 2.0f * G[i * KK + j];
            float pd = sqrtf(fmaxf(dsq, 0.0f));
            float t = fmaxf(TWO_DELTA_D - pd, 0.0f);
            psum += t * t * (pi * pres[j]);
        }
    }
    psum = waveSum32(psum);
    float npairs = ninst * (ninst - 1.0f) * 0.5f;
    float dist_b = psum / fmaxf(npairs, 1.0f);

    float validb = (ninst >= 2.0f) ? 1.0f : 0.0f;
    if (lane == 0) {
        atomicAdd(&accum[0], var_b * validb);
        atomicAdd(&accum[1], dist_b * validb);
        atomicAdd(&accum[2], reg_b * validb);
        atomicAdd(&accum[3], validb);
    }
}

// ---------------- kernel 5: final combine ----------------
__global__ void k_final(const float* __restrict__ accum, float* __restrict__ out) {
    if (threadIdx.x == 0 && blockIdx.x == 0) {
        float denom = fmaxf(accum[3], 1.0f);
        float v = accum[0] / denom;
        float di = accum[1] / denom;
        float rg = accum[2] / denom;
        out[0] = v + di + GAMMA_W * rg;   // ALPHA=BETA=1, GAMMA=0.001
        out[1] = v;
        out[2] = di;
        out[3] = rg;
    }
}

extern "C" void kernel_launch(void* const* d_in, const int* in_sizes, int n_in,
                              void* d_out, int out_size, void* d_ws, size_t ws_size,
                              hipStream_t stream) {
    (void)in_sizes; (void)n_in; (void)out_size; (void)ws_size;
    const float* emb = (const float*)d_in[0];
    const int* ids = (const int*)d_in[1];
    const unsigned char* mask = (const unsigned char*)d_in[2];
    float* ws = (float*)d_ws;
    float* out = (float*)d_out;

    float* sums = ws + WS_SUMS;     // becomes means after k_means
    float* counts = ws + WS_COUNTS;
    float* pen = ws + WS_PEN;
    float* accum = ws + WS_ACCUM;

    k_zero<<<(WS_FLOATS + 255) / 256, 256, 0, stream>>>(ws);
    k_segsum<<<dim3(BB, CHUNKS), 256, 0, stream>>>(emb, ids, mask, sums, counts);
    k_means<<<(BB * KK * DD + 255) / 256, 256, 0, stream>>>(sums, counts);
    k_var<<<dim3(BB, CHUNKS), 256, 0, stream>>>(emb, ids, mask, sums, pen);
    k_batch<<<BB, 32, 0, stream>>>(sums, counts, pen, accum);
    k_final<<<1, 32, 0, stream>>>(accum, out);
}